// LSForward_46050639348133
// MI455X (gfx1250) — compile-verified
//
#include <hip/hip_runtime.h>

namespace {

constexpr int   MQ    = 4096;
constexpr int   DOBS  = 64;
constexpr int   NIT   = 5;
constexpr int   JBS   = 32;          // j-split for Born matvec (parallelism)
constexpr int   JTILE = MQ / JBS;    // 128 j's per Born block
constexpr int   JSPL  = 64;          // j-split for far-field GEMM
constexpr float CG    = 0.07957747154594767f;  // 1/(4*pi)

typedef __attribute__((ext_vector_type(2))) float v2f;
typedef __attribute__((ext_vector_type(8))) float v8f;

// ---------------------------------------------------------------------------
// Init: VW = V*w ; incident wave p0(k) = exp(i k z)
// ---------------------------------------------------------------------------
__global__ void init_kernel(const float* __restrict__ V,
                            const float* __restrict__ pts,
                            const float* __restrict__ w,
                            float* __restrict__ VW,
                            float* __restrict__ p0) {
  int i = blockIdx.x * blockDim.x + threadIdx.x;
  if (i >= MQ) return;
  float vw = V[i] * w[i];
  VW[i] = vw;
  float dz = pts[i * 3 + 2];  // pts . dhat with dhat = (0,0,1)
  const float kv[4] = {0.5f, 1.0f, 1.5f, 2.0f};
#pragma unroll
  for (int k = 0; k < 4; ++k) {
    float sn, cs;
    __sincosf(kv[k] * dz, &sn, &cs);
    p0[(k * 2 + 0) * MQ + i] = cs;
    p0[(k * 2 + 1) * MQ + i] = sn;
  }
}

// ---------------------------------------------------------------------------
// Fused Born matvec partial: for one j-slab, accumulate
//   acc_r[k][i] = sum_j Gr(k)ij*ar(k)j - Gi(k)ij*ai(k)j     (and acc_i)
// G recomputed on the fly; e^{ikD} for all 4 k from ONE sincos(0.5*D)
// via exact double/compound-angle recurrence.
// grid = 16 row-blocks x 32 j-blocks, 256 threads (1 row each, 128 j's).
// ---------------------------------------------------------------------------
__global__ __launch_bounds__(256) void born_partial_kernel(
    const float* __restrict__ pts,
    const float* __restrict__ VW,
    const float* __restrict__ pin,
    float* __restrict__ part) {
  __shared__ float sx[JTILE], sy[JTILE], sz[JTILE], sr2[JTILE];
  __shared__ float sa[8][JTILE];  // [k*2+comp][j]  a = p*VW

  int tid   = threadIdx.x;
  int rb    = blockIdx.x & 15;
  int jb    = blockIdx.x >> 4;
  int i     = rb * 256 + tid;
  int jbase = jb * JTILE;

  // cooperative j-slab load (first JTILE threads)
  if (tid < JTILE) {
    int j = jbase + tid;
    float pxj = pts[j * 3 + 0], pyj = pts[j * 3 + 1], pzj = pts[j * 3 + 2];
    sx[tid] = pxj; sy[tid] = pyj; sz[tid] = pzj;
    sr2[tid] = pxj * pxj + pyj * pyj + pzj * pzj;
    float vwj = VW[j];
#pragma unroll
    for (int kc = 0; kc < 8; ++kc) sa[kc][tid] = pin[kc * MQ + j] * vwj;
  }
  __syncthreads();

  float xi = pts[i * 3 + 0], yi = pts[i * 3 + 1], zi = pts[i * 3 + 2];
  float r2i = xi * xi + yi * yi + zi * zi;

  float ar0 = 0.f, ai0 = 0.f, ar1 = 0.f, ai1 = 0.f;
  float ar2 = 0.f, ai2 = 0.f, ar3 = 0.f, ai3 = 0.f;

  for (int jj = 0; jj < JTILE; ++jj) {
    float dot = xi * sx[jj] + yi * sy[jj] + zi * sz[jj];
    float D2  = r2i + sr2[jj] - 2.0f * dot;
    D2 = fmaxf(D2, 1e-12f);
    float Dd = sqrtf(D2);
    float s  = CG / fmaxf(Dd, 1e-6f);
    s = (jbase + jj == i) ? 0.0f : s;     // zero diagonal (select, no branch)

    float s1, c1;
    __sincosf(0.5f * Dd, &s1, &c1);                        // k = 0.5
    float c2 = c1 * c1 - s1 * s1, s2 = 2.0f * c1 * s1;     // k = 1.0
    float c3 = c2 * c1 - s2 * s1, s3 = s2 * c1 + c2 * s1;  // k = 1.5
    float c4 = c2 * c2 - s2 * s2, s4 = 2.0f * c2 * s2;     // k = 2.0

    float br, bi;
    br = sa[0][jj]; bi = sa[1][jj];
    ar0 += s * (c1 * br - s1 * bi); ai0 += s * (c1 * bi + s1 * br);
    br = sa[2][jj]; bi = sa[3][jj];
    ar1 += s * (c2 * br - s2 * bi); ai1 += s * (c2 * bi + s2 * br);
    br = sa[4][jj]; bi = sa[5][jj];
    ar2 += s * (c3 * br - s3 * bi); ai2 += s * (c3 * bi + s3 * br);
    br = sa[6][jj]; bi = sa[7][jj];
    ar3 += s * (c4 * br - s4 * bi); ai3 += s * (c4 * bi + s4 * br);
  }

  float* pp = part + (size_t)jb * 8 * MQ;
  pp[0 * MQ + i] = ar0; pp[1 * MQ + i] = ai0;
  pp[2 * MQ + i] = ar1; pp[3 * MQ + i] = ai1;
  pp[4 * MQ + i] = ar2; pp[5 * MQ + i] = ai2;
  pp[6 * MQ + i] = ar3; pp[7 * MQ + i] = ai3;
}

// Deterministic fixed-order reduce of the j-slab partials:  p_out = p0 + sum
__global__ void born_reduce_kernel(const float* __restrict__ p0,
                                   const float* __restrict__ part,
                                   float* __restrict__ pout) {
  int idx = blockIdx.x * blockDim.x + threadIdx.x;  // over [8][MQ]
  if (idx >= 8 * MQ) return;
  int kc = idx / MQ;
  int i  = idx - kc * MQ;
  float acc = p0[idx];
#pragma unroll
  for (int jb = 0; jb < JBS; ++jb)
    acc += part[((size_t)jb * 8 + kc) * MQ + i];
  pout[idx] = acc;
}

// ---------------------------------------------------------------------------
// Far field via V_WMMA_F32_16X16X4_F32 (full f32 WMMA):
//   [2 x M] (w*V*psi re/im, rows 2..15 zero)  @  [M x 16] cos/sin phase tiles
// One wave per (k, 16-dir group, j-slab of 64). K stepped 4 at a time:
// each lane builds 2 B-phase values (ISA 4x16 B layout: lanes<16 -> K=0,1,
// lanes>=16 -> K=2,3) and the matching A fragment, where the row-select is a
// branch-free 0/1 mask FMA (keeps EXEC all-ones, no predicated LDS loads).
// Output lands in the documented C/D layout: VGPR0=row0, VGPR1=row1 for
// lanes 0..15 ->  fr = Pc[0]+Ps[1], fi = Pc[1]-Ps[0]  combined in-lane.
// ---------------------------------------------------------------------------
__global__ __launch_bounds__(32) void farfield_kernel(
    const float* __restrict__ pts,
    const float* __restrict__ V,
    const float* __restrict__ w,
    const float* __restrict__ obs,
    const float* __restrict__ pfin,
    float* __restrict__ ffp) {
  __shared__ float sxj[64], syj[64], szj[64], sazr[64], sazi[64];

  int lane = threadIdx.x;
  int b    = blockIdx.x;
  int jb   = b & (JSPL - 1);
  int dg   = (b >> 6) & 3;
  int ki   = b >> 8;
  float kk = 0.5f * (float)(ki + 1);
  int jbase = jb * (MQ / JSPL);  // 64 j's per block
  int d0 = dg * 16;

  for (int t = lane; t < 64; t += 32) {
    int j = jbase + t;
    sxj[t] = pts[j * 3 + 0];
    syj[t] = pts[j * 3 + 1];
    szj[t] = pts[j * 3 + 2];
    float wv = w[j] * V[j];
    sazr[t] = wv * pfin[(ki * 2 + 0) * MQ + j];
    sazi[t] = wv * pfin[(ki * 2 + 1) * MQ + j];
  }
  __syncthreads();

  int n   = lane & 15;          // B column / A row for this lane
  int off = (lane >> 4) << 1;   // lanes 0-15 -> K 0,1 ; lanes 16-31 -> K 2,3
  float ox = obs[(d0 + n) * 3 + 0];
  float oy = obs[(d0 + n) * 3 + 1];
  float oz = obs[(d0 + n) * 3 + 2];
  float m0 = (n == 0) ? 1.0f : 0.0f;   // row-0 mask (w*V*psi_re)
  float m1 = (n == 1) ? 1.0f : 0.0f;   // row-1 mask (w*V*psi_im)

  v8f accC = {0.f, 0.f, 0.f, 0.f, 0.f, 0.f, 0.f, 0.f};
  v8f accS = {0.f, 0.f, 0.f, 0.f, 0.f, 0.f, 0.f, 0.f};

  for (int j0 = 0; j0 < 64; j0 += 4) {
    int l0 = j0 + off, l1 = l0 + 1;
    float q0 = sxj[l0] * ox + syj[l0] * oy + szj[l0] * oz;
    float q1 = sxj[l1] * ox + syj[l1] * oy + szj[l1] * oz;
    float s0v, c0v, s1v, c1v;
    __sincosf(kk * q0, &s0v, &c0v);
    __sincosf(kk * q1, &s1v, &c1v);
    float r0 = sazr[l0], i0 = sazi[l0];
    float r1 = sazr[l1], i1 = sazi[l1];
    float a0 = fmaf(m0, r0, m1 * i0);   // branch-free A fragment
    float a1 = fmaf(m0, r1, m1 * i1);
    v2f af = {a0, a1};
    v2f bc = {c0v, c1v};
    v2f bs = {s0v, s1v};
    accC = __builtin_amdgcn_wmma_f32_16x16x4_f32(false, af, false, bc,
                                                 (short)0, accC, false, false);
    accS = __builtin_amdgcn_wmma_f32_16x16x4_f32(false, af, false, bs,
                                                 (short)0, accS, false, false);
  }

  if (lane < 16) {
    float fr = accC[0] + accS[1];  // sum w*zr*cos(kQ) + w*zi*sin(kQ)
    float fi = accC[1] - accS[0];  // sum w*zi*cos(kQ) - w*zr*sin(kQ)
    size_t base = ((((size_t)ki * 4 + dg) * JSPL) + jb) * 32 + (size_t)n * 2;
    ffp[base + 0] = fr;
    ffp[base + 1] = fi;
  }
}

// Deterministic fixed-order reduce over j-slabs, apply -C, write [4][64][2]
__global__ void ff_reduce_kernel(const float* __restrict__ ffp,
                                 float* __restrict__ out) {
  int idx = blockIdx.x * blockDim.x + threadIdx.x;
  if (idx >= 4 * DOBS * 2) return;
  int ki = idx / (DOBS * 2);
  int r  = idx - ki * DOBS * 2;
  int d  = r >> 1;
  int comp = r & 1;
  int dg = d >> 4;
  int n  = d & 15;
  float acc = 0.0f;
  for (int jb = 0; jb < JSPL; ++jb)
    acc += ffp[((((size_t)ki * 4 + dg) * JSPL) + jb) * 32 + (size_t)n * 2 + comp];
  out[idx] = -CG * acc;
}

}  // namespace

extern "C" void kernel_launch(void* const* d_in, const int* in_sizes, int n_in,
                              void* d_out, int out_size, void* d_ws, size_t ws_size,
                              hipStream_t stream) {
  (void)in_sizes; (void)n_in; (void)out_size;
  const float* V   = (const float*)d_in[0];
  const float* obs = (const float*)d_in[1];
  const float* pts = (const float*)d_in[2];
  const float* w   = (const float*)d_in[3];
  float* ws = (float*)d_ws;

  // workspace layout (floats)
  float* VW   = ws;                               // MQ
  float* p0   = VW + MQ;                          // 8*MQ
  float* pA   = p0 + 8 * MQ;                      // 8*MQ
  float* pB   = pA + 8 * MQ;                      // 8*MQ
  float* part = pB + 8 * MQ;                      // JBS*8*MQ
  float* ffp  = part + (size_t)JBS * 8 * MQ;      // 4*4*JSPL*32
  size_t need_floats = (size_t)(ffp - ws) + (size_t)4 * 4 * JSPL * 32;
  if (ws_size < need_floats * sizeof(float)) return;

  init_kernel<<<MQ / 256, 256, 0, stream>>>(V, pts, w, VW, p0);

  // it0: p0 -> pA ; it1: pA -> pB ; it2: pB -> pA ; it3: pA -> pB ; it4: pB -> pA
  float* pin  = p0;
  float* pout = pA;
  for (int it = 0; it < NIT; ++it) {
    born_partial_kernel<<<16 * JBS, 256, 0, stream>>>(pts, VW, pin, part);
    born_reduce_kernel<<<(8 * MQ) / 256, 256, 0, stream>>>(p0, part, pout);
    pin  = pout;
    pout = (pout == pA) ? pB : pA;
  }
  // final field is in pA (NIT = 5)

  farfield_kernel<<<4 * 4 * JSPL, 32, 0, stream>>>(pts, V, w, obs, pin, ffp);
  ff_reduce_kernel<<<2, 256, 0, stream>>>(ffp, (float*)d_out);
}